// LSG_output_7069516169615
// MI455X (gfx1250) — compile-verified
//
#include <hip/hip_runtime.h>

// ---------------------------------------------------------------------------
// Fused exp_convolve (linear recurrence over T) + f32 GEMM (btj,jk->btk)
// via CDNA5 V_WMMA_F32_16X16X4_F32 (full f32 precision matrix pipe).
//
// Shapes: B=32, T=2000, J=512, N=128, f32 everywhere.
// Three passes (blocked parallel scan over T):
//   1) per (b,chunk): zero-init local scan, keep last state (tail)
//   2) per (b,j):     carry[k+1] = d^80 * carry[k] + tail[k]   (25 steps)
//   3) per (b,chunk): resume scan from carry; every 16 timesteps stage the
//      16x512 filtered tile in LDS and run 8 waves x 128 wmma_f32_16x16x4
//      against W (pre-staged transposed in LDS, padded stride 516 => the
//      half-wave ds_load_2addr_b64 operand fetches are bank-conflict free).
//   z (131 MB) < L2 (192 MB) so pass 3's z re-read is an L2 hit; HBM traffic
//   is ~131 MB in + 33 MB out ~= the 7 us roofline floor at 23.3 TB/s.
//
// This revision: (a) two interleaved WMMA accumulators to halve the D->C
// dependent-chain depth, (b) software-pipelined z prefetch so the next
// tile's global loads overlap the current tile's 128-WMMA K-loop.
// ---------------------------------------------------------------------------

typedef __attribute__((ext_vector_type(2))) float v2f;
typedef __attribute__((ext_vector_type(8))) float v8f;

#define BB      32
#define TT_TOT  2000
#define JJ      512
#define NN      128
#define CHUNK   80      // timesteps per workgroup-chunk (5 tiles of 16)
#define NCHUNK  25      // 2000 / 80
#define TILE_T  16
#define NTILE   5       // CHUNK / TILE_T
#define WSTRIDE 516     // padded LDS row stride (dwords); 4*dn==2 mod 64 unsolvable
                        // => the two half-wave bank sets are disjoint -> no conflicts

// ---------------- Pass 1: per-chunk zero-init scan, keep tail --------------
__global__ __launch_bounds__(256) void lsg_chunk_tail(
    const float* __restrict__ z, const float* __restrict__ decay,
    float* __restrict__ tail) {
  const int b = blockIdx.x / NCHUNK;
  const int c = blockIdx.x % NCHUNK;
  const int tid = threadIdx.x;
  const int j0 = tid, j1 = tid + 256;
  const float d0 = decay[j0], d1 = decay[j1];
  float s0 = 0.f, s1 = 0.f;
  const float* zp = z + ((long)b * TT_TOT + (long)c * CHUNK) * JJ;
#pragma unroll 8
  for (int t = 0; t < CHUNK; ++t) {
    s0 = d0 * s0 + (1.f - d0) * zp[t * JJ + j0];
    s1 = d1 * s1 + (1.f - d1) * zp[t * JJ + j1];
  }
  tail[(b * NCHUNK + c) * JJ + j0] = s0;
  tail[(b * NCHUNK + c) * JJ + j1] = s1;
}

// ---------------- Pass 2: sequential carry across the 25 chunks ------------
__global__ __launch_bounds__(256) void lsg_carry_scan(
    const float* __restrict__ decay, const float* __restrict__ tail,
    float* __restrict__ carry) {
  const int g = blockIdx.x * 256 + threadIdx.x;  // b*512 + j
  const int b = g >> 9;
  const int j = g & (JJ - 1);
  const float d = decay[j];
  // d^80 = d^64 * d^16 via exact repeated squaring
  const float d2 = d * d, d4 = d2 * d2, d8 = d4 * d4, d16 = d8 * d8;
  const float d32 = d16 * d16, d64 = d32 * d32;
  const float dC = d64 * d16;
  float c = 0.f;
  for (int k = 0; k < NCHUNK; ++k) {
    carry[(b * NCHUNK + k) * JJ + j] = c;
    c = dC * c + tail[(b * NCHUNK + k) * JJ + j];
  }
}

// ---------------- Pass 3: fused scan + WMMA GEMM ---------------------------
__global__ __launch_bounds__(256) void lsg_scan_wmma(
    const float* __restrict__ z, const float* __restrict__ decay,
    const float* __restrict__ W, const float* __restrict__ carry,
    float* __restrict__ out) {
  // LDS: W transposed [n][k] (128 x 516) + filtered tile [tt][j] (16 x 516)
  __shared__ float lds[(NN + TILE_T) * WSTRIDE];  // 74304 dwords = 297,216 B
  float* sW = lds;                  // sW[n*WSTRIDE + k] = W[k*NN + n]
  float* sF = lds + NN * WSTRIDE;   // sF[tt*WSTRIDE + j]

  const int b = blockIdx.x / NCHUNK;
  const int c = blockIdx.x % NCHUNK;
  const int tid = threadIdx.x;

  // Stage W transposed into LDS (one-time, served from L2 after first WG).
#pragma unroll 4
  for (int it = 0; it < (JJ * NN) / 256; ++it) {
    const int idx = it * 256 + tid;
    const int k = idx >> 7;        // row j of W
    const int n = idx & (NN - 1);  // col n of W
    sW[n * WSTRIDE + k] = W[idx];
  }

  // Scan state: each thread owns channels j0 and j1.
  const int j0 = tid, j1 = tid + 256;
  const float d0 = decay[j0], d1 = decay[j1];
  float s0 = carry[(b * NCHUNK + c) * JJ + j0];
  float s1 = carry[(b * NCHUNK + c) * JJ + j1];

  // WMMA operand addressing (wave32):
  //  A (16x4 f32): lanes 0-15 hold row M=l16, K = k0+{0,1}; lanes 16-31 K = k0+{2,3}
  //  B (4x16 f32): lanes 0-15 hold col N=l16, K = k0+{0,1}; lanes 16-31 K = k0+{2,3}
  const int lane = tid & 31;
  const int wv   = tid >> 5;            // wave id 0..7 -> output cols [16*wv, 16*wv+16)
  const int h2   = ((lane >> 4) & 1) << 1;  // 0 or 2
  const int l16  = lane & 15;
  const int aBase = l16 * WSTRIDE + h2;              // sF row = M, col = k
  const int bBase = (wv * 16 + l16) * WSTRIDE + h2;  // sW row = n, col = k

  const float* zp = z + ((long)b * TT_TOT + (long)c * CHUNK) * JJ;
  float* op = out + ((long)b * TT_TOT + (long)c * CHUNK) * NN;

  // Software-pipelined z tile prefetch (double-buffered registers).
  float za[2][TILE_T], zb[2][TILE_T];
#pragma unroll
  for (int t = 0; t < TILE_T; ++t) {
    za[0][t] = zp[t * JJ + j0];
    zb[0][t] = zp[t * JJ + j1];
  }

  __syncthreads();  // sW staged

  for (int tile = 0; tile < NTILE; ++tile) {
    const int cur = tile & 1, nxt = cur ^ 1;

    // Sequential exp-convolve on prefetched tile; stage filtered tile in LDS.
#pragma unroll
    for (int t = 0; t < TILE_T; ++t) {
      s0 = d0 * s0 + (1.f - d0) * za[cur][t];
      s1 = d1 * s1 + (1.f - d1) * zb[cur][t];
      sF[t * WSTRIDE + j0] = s0;
      sF[t * WSTRIDE + j1] = s1;
    }
    __syncthreads();

    // Issue next tile's global loads now so they overlap the WMMA K-loop.
    if (tile + 1 < NTILE) {
#pragma unroll
      for (int t = 0; t < TILE_T; ++t) {
        za[nxt][t] = zp[((tile + 1) * TILE_T + t) * JJ + j0];
        zb[nxt][t] = zp[((tile + 1) * TILE_T + t) * JJ + j1];
      }
    }

    // 16x16 output tile per wave, K=512 -> 128 wmma f32 on two interleaved
    // accumulators (halves the D->C dependent-chain depth).
    v8f acc0 = {}, acc1 = {};
#pragma unroll 4
    for (int k0 = 0; k0 < JJ; k0 += 8) {
      const v2f a0 = *(const v2f*)(sF + aBase + k0);      // ds_load_2addr_b64
      const v2f b0 = *(const v2f*)(sW + bBase + k0);      //   pairs with +4
      const v2f a1 = *(const v2f*)(sF + aBase + k0 + 4);
      const v2f b1 = *(const v2f*)(sW + bBase + k0 + 4);
      acc0 = __builtin_amdgcn_wmma_f32_16x16x4_f32(
          false, a0, false, b0, (short)0, acc0, false, false);
      acc1 = __builtin_amdgcn_wmma_f32_16x16x4_f32(
          false, a1, false, b1, (short)0, acc1, false, false);
    }

    // D layout: VGPR r -> row M = r + 8*(lane>=16), col N = l16 (+16*wv)
    const int mAdd = (h2 >> 1) * 8;  // 0 or 8
    const int col  = wv * 16 + l16;
#pragma unroll
    for (int r = 0; r < 8; ++r) {
      const int t = tile * TILE_T + r + mAdd;
      op[t * NN + col] = acc0[r] + acc1[r];
    }
    __syncthreads();  // protect sF before next tile's scan overwrites it
  }
}

// ---------------------------------------------------------------------------
extern "C" void kernel_launch(void* const* d_in, const int* in_sizes, int n_in,
                              void* d_out, int out_size, void* d_ws, size_t ws_size,
                              hipStream_t stream) {
  const float* z     = (const float*)d_in[0];  // [32, 2000, 512]
  const float* decay = (const float*)d_in[1];  // [512]
  const float* W     = (const float*)d_in[2];  // [512, 128]
  float* out = (float*)d_out;                  // [32, 2000, 128]

  float* tail  = (float*)d_ws;                 // [32, 25, 512]
  float* carry = tail + (size_t)BB * NCHUNK * JJ;

  lsg_chunk_tail<<<dim3(BB * NCHUNK), dim3(256), 0, stream>>>(z, decay, tail);
  lsg_carry_scan<<<dim3((BB * JJ) / 256), dim3(256), 0, stream>>>(decay, tail, carry);
  lsg_scan_wmma<<<dim3(BB * NCHUNK), dim3(256), 0, stream>>>(z, decay, W, carry, out);
}